// PoseNoiseTransform_68135361183838
// MI455X (gfx1250) — compile-verified
//
#include <hip/hip_runtime.h>
#include <cstdint>
#include <cmath>

// Problem constants (from reference: ts=2048, ns=4096)
#define TS      2048
#define NSQ     4096
#define STEPS   (TS - 1)          // 2047
#define CSZ     64                // time steps per chunk (last chunk: 63)
#define NCHUNK  ((STEPS + CSZ - 1) / CSZ)   // 32
#define STDF    0.05f
#define DEPTH   4                 // async pipeline depth (ring slots)

typedef float vq __attribute__((ext_vector_type(4)));   // (w, x, y, z)

__device__ __forceinline__ vq qmul(vq a, vq b) {
    vq r;
    r.x = a.x * b.x - a.y * b.y - a.z * b.z - a.w * b.w;
    r.y = a.x * b.y + a.y * b.x + a.z * b.w - a.w * b.z;
    r.z = a.x * b.z - a.y * b.w + a.z * b.x + a.w * b.y;
    r.w = a.x * b.w + a.y * b.z - a.z * b.y + a.w * b.x;
    return r;
}

// CDNA5 async copy: per-lane 16B global -> LDS, tracked by ASYNCcnt.
// GVS addressing: 64-bit SGPR base + 32-bit per-lane byte offset.
#define ASYNC_B128(ldsoff, voff, sbase)                                     \
    asm volatile("global_load_async_to_lds_b128 %0, %1, %2"                 \
                 :: "v"(ldsoff), "v"(voff), "s"(sbase) : "memory")
#define WAIT_ASYNC(n)                                                       \
    asm volatile("s_wait_asynccnt " #n ::: "memory")

// Phase 1 (FINAL=false): per (chunk, column) accumulate chunk-local
//   prefix product of q_dot, write aggregate to ws.
// Phase 3 (FINAL=true): recompute local prefix, multiply by precomputed
//   base[c][n] = exclusive-chunk-prefix * q0, store output rows 1..2047.
template <bool FINAL>
__global__ __launch_bounds__(256) void pose_chunk(
    const float* __restrict__ qg, const float* __restrict__ ng,
    const float* __restrict__ baseg, float* __restrict__ outg)
{
    // Typed LDS ring buffers -> direct ds_load_b128 on the read side.
    __shared__ __align__(16) vq sQ[DEPTH][256];   // q[t+1] rows
    __shared__ __align__(16) vq sN[DEPTH][256];   // noise[t] rows

    const int tid = threadIdx.x;
    const int n   = blockIdx.x * 256 + tid;   // column
    const int c   = blockIdx.y;               // time chunk
    const int t0  = c * CSZ;
    const int t1  = (t0 + CSZ < STEPS) ? (t0 + CSZ) : STEPS;

    const uint32_t rowB = (uint32_t)NSQ * 16u;          // 64 KiB per time row
    const uint32_t colB = (uint32_t)n * 16u;

    // q[t0] straight to registers (consumed immediately).
    vq qt = *(const vq*)(qg + ((size_t)t0 * NSQ + n) * 4);

    vq acc = (vq){1.f, 0.f, 0.f, 0.f};
    vq bq;
    if (FINAL) bq = *(const vq*)(baseg + ((size_t)c * NSQ + n) * 4);

    // Prologue: stage steps t0 .. t0+DEPTH-2 into ring slots 0..DEPTH-2.
    uint32_t voffQ = (uint32_t)(t0 + 1) * rowB + colB;
    uint32_t voffN = (uint32_t)t0 * rowB + colB;
#pragma unroll
    for (int p = 0; p < DEPTH - 1; ++p) {
        ASYNC_B128((uint32_t)(size_t)&sQ[p][tid], voffQ, qg); voffQ += rowB;
        ASYNC_B128((uint32_t)(size_t)&sN[p][tid], voffN, ng); voffN += rowB;
    }

    // One scan step consuming ring slot `slot` for time index t.
    int t = t0;
    auto step = [&](int slot) {
        vq qn = sQ[slot][tid];          // q[t+1]  (ds_load_b128)
        vq nz = sN[slot][tid];          // noise[t]

        // q_dot = q[t+1] * conj(q[t])
        vq qc = (vq){qt.x, -qt.y, -qt.z, -qt.w};
        vq qd = qmul(qn, qc);
        // q_samp = normalize((noise-0.5) * (1, STD, STD, STD))
        vq s = (vq){ nz.x - 0.5f, (nz.y - 0.5f) * STDF,
                     (nz.z - 0.5f) * STDF, (nz.w - 0.5f) * STDF };
        float inv = 1.0f / sqrtf(s.x*s.x + s.y*s.y + s.z*s.z + s.w*s.w);
        qd = qmul(qd, s * inv);
        // inclusive prefix, newest on the left
        acc = qmul(qd, acc);

        if (FINAL) {
            vq r = qmul(acc, bq);
            __builtin_nontemporal_store(
                r, (vq*)(outg + ((size_t)(t + 1) * NSQ + n) * 4));
        }
        qt = qn;
    };

    // Main loop: keep DEPTH-1 steps in flight; async loads complete in
    // order, so ASYNCcnt <= 2*(DEPTH-1) guarantees step t's pair landed.
    for (; t + (DEPTH - 1) < t1; ++t) {
        const int islot = (t - t0 + DEPTH - 1) & (DEPTH - 1);
        ASYNC_B128((uint32_t)(size_t)&sQ[islot][tid], voffQ, qg); voffQ += rowB;
        ASYNC_B128((uint32_t)(size_t)&sN[islot][tid], voffN, ng); voffN += rowB;
        WAIT_ASYNC(6);
        step((t - t0) & (DEPTH - 1));
    }
    // Epilogue: exactly DEPTH-1 = 3 steps remain (chunk length >= 63).
    WAIT_ASYNC(4); step((t - t0) & (DEPTH - 1)); ++t;
    WAIT_ASYNC(2); step((t - t0) & (DEPTH - 1)); ++t;
    WAIT_ASYNC(0); step((t - t0) & (DEPTH - 1)); ++t;

    if (!FINAL)
        *(vq*)(outg + ((size_t)c * NSQ + n) * 4) = acc;   // chunk aggregate
}

// Phase 2: per column, exclusive scan over the 32 chunk aggregates,
// folded with q0; also emits output row 0 (= q[0]).
__global__ __launch_bounds__(256) void pose_scan(
    const float* __restrict__ qg, const float* __restrict__ aggg,
    float* __restrict__ baseg, float* __restrict__ outg)
{
    const int n = blockIdx.x * 256 + threadIdx.x;
    vq q0 = *(const vq*)(qg + (size_t)n * 4);
    *(vq*)(outg + (size_t)n * 4) = q0;                    // out[0] = q[0]
    vq run = (vq){1.f, 0.f, 0.f, 0.f};                    // exc[0] = identity
    for (int c = 0; c < NCHUNK; ++c) {
        *(vq*)(baseg + ((size_t)c * NSQ + n) * 4) = qmul(run, q0);
        run = qmul(*(const vq*)(aggg + ((size_t)c * NSQ + n) * 4), run);
    }
}

extern "C" void kernel_launch(void* const* d_in, const int* in_sizes, int n_in,
                              void* d_out, int out_size, void* d_ws, size_t ws_size,
                              hipStream_t stream)
{
    (void)in_sizes; (void)n_in; (void)out_size; (void)ws_size;
    const float* qg   = (const float*)d_in[0];   // [2048, 4096, 4] f32
    const float* ng   = (const float*)d_in[1];   // [2047, 4096, 4] f32
    float*       outg = (float*)d_out;           // [2048, 4096, 4] f32
    float*       aggg = (float*)d_ws;                         // 2 MiB
    float*       bseg = aggg + (size_t)NCHUNK * NSQ * 4;      // 2 MiB

    dim3 blk(256, 1, 1);
    dim3 grid(NSQ / 256, NCHUNK, 1);     // 16 x 32 blocks, 4096 waves

    pose_chunk<false><<<grid, blk, 0, stream>>>(qg, ng, nullptr, aggg);
    pose_scan<<<dim3(NSQ / 256, 1, 1), blk, 0, stream>>>(qg, aggg, bseg, outg);
    pose_chunk<true><<<grid, blk, 0, stream>>>(qg, ng, bseg, outg);
}